// OhemCrossEntropy2d_m_32229434589492
// MI455X (gfx1250) — compile-verified
//
#include <hip/hip_runtime.h>
#include <cstdint>

// ---------------------------------------------------------------------------
// OHEM CrossEntropy2d for MI455X (gfx1250, wave32).
//
// Memory-bound: ~350 MB streamed once -> ~15 us floor at 23.3 TB/s.
// No matmul exists in this op, so WMMA is not applicable; the CDNA5 path that
// pays off is the Tensor Data Mover: one 2-D descriptor per 1024-pixel tile
// gathers all 19 channel rows (2 MB apart in [n,c,h,w]) into contiguous LDS,
// double-buffered and tracked with TENSORcnt, overlapping DMA with compute.
// Each thread drains 4 contiguous pixels via ds_load_b128.
// ---------------------------------------------------------------------------

#define N_IMG   8
#define C_CLS   19
#define HW      (512 * 1024)          // pixels per image
#define TILE    1024                  // pixels per TDM tile
#define N_TILES ((N_IMG * HW) / TILE) // 4096
#define TPI     (HW / TILE)           // tiles per image = 512
#define IGNORE_LBL 255
#define TH1 0.8f
#define TH2 0.5f

typedef __attribute__((ext_vector_type(4))) unsigned int v4u;
typedef __attribute__((ext_vector_type(4))) int          v4i;
typedef __attribute__((ext_vector_type(8))) int          v8i;
typedef __attribute__((ext_vector_type(4))) float        v4f;

// Issue a TDM load of one tile: channel c, pixel hw0 + p of image n in predict goes to LDS row c, column p,
// p in [0,TILE), c in [0,19).  D# per CDNA5 ISA.
__device__ __forceinline__ void tdm_issue_tile(const float* __restrict__ predict,
                                               int tile, unsigned lds_byte_off) {
    const int      n    = tile >> 9;                // tile / TPI
    const int      hw0  = (tile & (TPI - 1)) << 10; // (tile % TPI) * TILE
    const uint64_t ga   = (uint64_t)(uintptr_t)predict +
                          4ull * ((uint64_t)n * (uint64_t)(C_CLS * HW) + (unsigned)hw0);

    v4u g0;
    g0[0] = 1u;                                     // count=1, user descriptor
    g0[1] = lds_byte_off;                           // lds_addr
    g0[2] = (unsigned)(ga & 0xFFFFFFFFu);           // global_addr[31:0]
    g0[3] = (unsigned)((ga >> 32) & 0x01FFFFFFu)    // global_addr[56:32]
          | 0x80000000u;                            // type = 2 ("image")

    v8i g1;
    g1[0] = 0x00020000;                 // wg_mask=0, data_size=2 (4 bytes)
    g1[1] = 0;                          // no atomic barrier; tensor_dim0[15:0]=0
    g1[2] = 0x8 | (C_CLS << 16);        // tensor_dim0=0x80000 (hi16) | tensor_dim1=19
    g1[3] = (TILE << 16);               // tensor_dim1 hi16=0 | tile_dim0=1024
    g1[4] = C_CLS;                      // tile_dim1=19, tile_dim2=0
    g1[5] = HW;                         // tensor_dim0_stride lo32 = 524288
    g1[6] = 0;                          // stride hi16 = 0 | tensor_dim1_stride lo16
    g1[7] = 0;

    v4i g2 = {0, 0, 0, 0};              // 2-D tensor: groups 2/3 unused
    v4i g3 = {0, 0, 0, 0};

#if defined(__clang_major__) && (__clang_major__ >= 23)
    v8i g4 = {0, 0, 0, 0, 0, 0, 0, 0};
    __builtin_amdgcn_tensor_load_to_lds(g0, g1, g2, g3, g4, 0);
#else
    __builtin_amdgcn_tensor_load_to_lds(g0, g1, g2, g3, 0);
#endif
}

__device__ __forceinline__ float wave_reduce_sum(float v) {
    // wave32 butterfly
    v += __shfl_down(v, 16, 32);
    v += __shfl_down(v,  8, 32);
    v += __shfl_down(v,  4, 32);
    v += __shfl_down(v,  2, 32);
    v += __shfl_down(v,  1, 32);
    return v;
}

__global__ void ohem_init_kernel(double* __restrict__ acc) {
    if (threadIdx.x < 6) acc[threadIdx.x] = 0.0;
}

__global__ __launch_bounds__(256)
void ohem_main_kernel(const float* __restrict__ predict,
                      const int*   __restrict__ target,
                      const float* __restrict__ class_weight,
                      double*      __restrict__ acc) {
    __shared__ float buf[2][C_CLS * TILE];   // 2 x 76 KB double buffer
    __shared__ float cw_s[C_CLS];
    __shared__ float acc_s[6];               // den_e,num_e,den_m,num_m,den_h,num_h

    const int tid = threadIdx.x;
    if (tid < C_CLS) cw_s[tid] = class_weight[tid];
    if (tid < 6)     acc_s[tid] = 0.0f;

    const unsigned lds_off0 = (unsigned)(uintptr_t)(const void*)&buf[0][0];
    const unsigned lds_off1 = (unsigned)(uintptr_t)(const void*)&buf[1][0];

    float den_e = 0.f, num_e = 0.f, den_m = 0.f, num_m = 0.f, den_h = 0.f, num_h = 0.f;

    const int t0     = blockIdx.x;
    const int stride = gridDim.x;

    // Prologue: issue first tile into buffer 0 (lane 0 of wave 0 only;
    // other waves skip via execz branch).
    if (tid == 0 && t0 < N_TILES) tdm_issue_tile(predict, t0, lds_off0);

    int parity = 0;
    for (int t = t0; t < N_TILES; t += stride) {
        const int tn = t + stride;
        if (tid == 0) {
            if (tn < N_TILES) {
                tdm_issue_tile(predict, tn, parity ? lds_off0 : lds_off1);
                __builtin_amdgcn_s_wait_tensorcnt(1);  // current tile resident
            } else {
                __builtin_amdgcn_s_wait_tensorcnt(0);
            }
        }
        __syncthreads();  // publish buf[parity] to all 8 waves

        const float* __restrict__ src = &buf[parity][0];
        const int n     = t >> 9;
        const int pbase = n * HW + ((t & (TPI - 1)) << 10);
        const int p0    = tid * 4;            // this thread's 4 contiguous pixels

        // 4 targets with one non-temporal b128 load
        const v4i tg = __builtin_nontemporal_load((const v4i*)&target[pbase + p0]);

        // 19 channel rows, ds_load_b128 each (consecutive lanes -> consecutive
        // 16B, conflict-free)
        v4f xv[C_CLS];
        #pragma unroll
        for (int c = 0; c < C_CLS; ++c)
            xv[c] = *(const v4f*)(src + c * TILE + p0);

        v4f m = xv[0];
        #pragma unroll
        for (int c = 1; c < C_CLS; ++c)
            m = __builtin_elementwise_max(m, xv[c]);

        v4f s = {0.f, 0.f, 0.f, 0.f};
        #pragma unroll
        for (int c = 0; c < C_CLS; ++c) {     // v_exp_f32 x4 per channel
            s[0] += __expf(xv[c][0] - m[0]);
            s[1] += __expf(xv[c][1] - m[1]);
            s[2] += __expf(xv[c][2] - m[2]);
            s[3] += __expf(xv[c][3] - m[3]);
        }

        #pragma unroll
        for (int j = 0; j < 4; ++j) {
            const float lse = m[j] + __logf(s[j]);
            const int   tgt = tg[j];
            const bool  valid = (tgt != IGNORE_LBL);
            const int   ts    = valid ? tgt : 0;
            const float xt  = src[ts * TILE + p0 + j];  // LDS: no dyn reg index
            const float nll = lse - xt;
            const float pt  = __expf(xt - lse);         // true-class prob
            const float wt  = valid ? cw_s[ts] : 0.0f;

            const float we = (pt >= TH1)              ? wt : 0.f;
            const float wm = (pt <  TH1 && pt >= TH2) ? wt : 0.f;
            const float wh = (pt <  TH2)              ? wt : 0.f;
            den_e += we; num_e += we * nll;
            den_m += wm; num_m += wm * nll;
            den_h += wh; num_h += wh * nll;
        }

        __syncthreads();  // all reads of buf[parity] done before it is re-DMA'd
        parity ^= 1;
    }

    // wave32 reduction -> per-wave ds_add_f32 -> per-block f64 global atomics
    float r[6] = {den_e, num_e, den_m, num_m, den_h, num_h};
    #pragma unroll
    for (int k = 0; k < 6; ++k) {
        const float w = wave_reduce_sum(r[k]);
        if ((tid & 31) == 0) atomicAdd(&acc_s[k], w);
    }
    __syncthreads();
    if (tid < 6) atomicAdd(&acc[tid], (double)acc_s[tid]);  // global_atomic_add_f64
}

__global__ void ohem_finalize_kernel(const double* __restrict__ acc,
                                     float* __restrict__ out) {
    double loss = 0.0;
    #pragma unroll
    for (int k = 0; k < 3; ++k) {
        const double den = acc[2 * k];
        const double num = acc[2 * k + 1];
        loss += num / fmax(den, 1e-12);
    }
    out[0] = (float)loss;
}

extern "C" void kernel_launch(void* const* d_in, const int* in_sizes, int n_in,
                              void* d_out, int out_size, void* d_ws, size_t ws_size,
                              hipStream_t stream) {
    const float* predict      = (const float*)d_in[0];  // [8,19,512,1024] f32
    const int*   target       = (const int*)  d_in[1];  // [8,512,1024] int
    const float* class_weight = (const float*)d_in[2];  // [19] f32
    float*       out          = (float*)d_out;          // scalar f32
    double*      acc          = (double*)d_ws;          // 6 f64 accumulators

    (void)in_sizes; (void)n_in; (void)out_size; (void)ws_size;

    ohem_init_kernel<<<1, 32, 0, stream>>>(acc);
    ohem_main_kernel<<<1024, 256, 0, stream>>>(predict, target, class_weight, acc);
    ohem_finalize_kernel<<<1, 1, 0, stream>>>(acc, out);
}